// ResidualBlock_75419625718609
// MI455X (gfx1250) — compile-verified
//
#include <hip/hip_runtime.h>

// CDNA5 / gfx1250 — wave32, WMMA f32 16x16x4.
typedef float v2f __attribute__((ext_vector_type(2)));
typedef float v8f __attribute__((ext_vector_type(8)));

#define SLOPE 0.2f

__device__ __forceinline__ float leaky(float v) {
    return v >= 0.f ? v : SLOPE * v;
}

// D = A(16x4, f32) * B(4x16, f32) + C(16x16, f32)
__device__ __forceinline__ v8f wmma_f32_4(v2f a, v2f b, v8f c) {
    return __builtin_amdgcn_wmma_f32_16x16x4_f32(
        /*neg_a=*/false, a, /*neg_b=*/false, b,
        /*c_mod=*/(short)0, c, /*reuse_a=*/false, /*reuse_b=*/false);
}

// Packed-weight element counts (v2f elements)
#define NFRONT (20 * 64 * 32)   // Wres(16 tiles)+Winit(4 tiles), 64 kk steps
#define NFIN   (16 * 64 * 32)   // Wfin, 16 tiles, 64 kk steps
#define NG1    (4 * 32)         // Wg1, 4 tiles, K=4 (1 step)
#define NG2    (8 * 32)         // Wg2, 8 tiles

// ---------------------------------------------------------------------------
// Pack kernel: rearrange weights into per-lane B-fragment order so GEMM loads
// are one contiguous global_load_b64 per lane:
//   P[((t*KK + kk)*32 + lane)] = { W[k0][n], W[k0+1][n] },
//   k0 = kk*4 + (lane>>4)*2,  n = t*16 + (lane&15)
// ---------------------------------------------------------------------------
__global__ __launch_bounds__(256) void k_pack(
    const float* __restrict__ Wres, const float* __restrict__ Winit,
    const float* __restrict__ Wg1,  const float* __restrict__ Wg2,
    const float* __restrict__ Wfin,
    float* __restrict__ Pfront, float* __restrict__ Pg1,
    float* __restrict__ Pg2,    float* __restrict__ Pfin)
{
    int tid = blockIdx.x * 256 + threadIdx.x;
    if (tid < NFRONT) {
        int e = tid;
        int lane = e & 31, kk = (e >> 5) & 63, t = e >> 11;
        int l = lane & 15, hl = lane >> 4;
        int k0 = kk * 4 + hl * 2;
        const float* W; int outd, n;
        if (t < 16) { W = Wres;  outd = 256; n = t * 16 + l; }
        else        { W = Winit; outd = 64;  n = (t - 16) * 16 + l; }
        Pfront[e * 2 + 0] = W[(k0 + 0) * outd + n];
        Pfront[e * 2 + 1] = W[(k0 + 1) * outd + n];
    } else if (tid < NFRONT + NFIN) {
        int e = tid - NFRONT;
        int lane = e & 31, kk = (e >> 5) & 63, t = e >> 11;
        int l = lane & 15, hl = lane >> 4;
        int k0 = kk * 4 + hl * 2;
        int n = t * 16 + l;
        Pfin[e * 2 + 0] = Wfin[(k0 + 0) * 256 + n];
        Pfin[e * 2 + 1] = Wfin[(k0 + 1) * 256 + n];
    } else if (tid < NFRONT + NFIN + NG1) {
        int e = tid - NFRONT - NFIN;
        int lane = e & 31, t = e >> 5;
        int l = lane & 15, hl = lane >> 4;
        int n = t * 16 + l;
        Pg1[e * 2 + 0] = Wg1[(hl * 2 + 0) * 64 + n];
        Pg1[e * 2 + 1] = Wg1[(hl * 2 + 1) * 64 + n];
    } else if (tid < NFRONT + NFIN + NG1 + NG2) {
        int e = tid - NFRONT - NFIN - NG1;
        int lane = e & 31, t = e >> 5;
        int l = lane & 15, hl = lane >> 4;
        int n = t * 16 + l;
        Pg2[e * 2 + 0] = Wg2[(hl * 2 + 0) * 128 + n];
        Pg2[e * 2 + 1] = Wg2[(hl * 2 + 1) * 128 + n];
    }
}

// ---------------------------------------------------------------------------
// Kernel 1: residual = leaky(X @ Wres + bres)  [N,256]
//           a0       = leaky(X @ Winit + binit) [N,64]
// Workgroup (4 waves) per 16-row tile; 20 col tiles split 5/wave. X tile in
// LDS (stride 258 -> conflict-free ds_load_b64). B-fragments from packed
// weights: one coalesced b64 per lane per WMMA, int offsets only.
// ---------------------------------------------------------------------------
__global__ __launch_bounds__(128) void k_front(
    const float* __restrict__ X,
    const float* __restrict__ Pfront,
    const float* __restrict__ Bres, const float* __restrict__ Binit,
    float* __restrict__ res_ws, float* __restrict__ a0_ws, int N)
{
    __shared__ float xs[16 * 258];
    const int tid = threadIdx.x;
    const int m0  = blockIdx.x * 16;

    {   // Stage 16x256 X tile (coalesced float2 loads, int offsets).
        const float2* src = (const float2*)X + m0 * 128;
        for (int p = tid; p < 16 * 128; p += 128) {
            int row = p >> 7, c2 = p & 127;
            float2 v = src[row * 128 + c2];
            xs[row * 258 + c2 * 2 + 0] = v.x;
            xs[row * 258 + c2 * 2 + 1] = v.y;
        }
    }
    __syncthreads();

    const int lane = tid & 31, wv = tid >> 5;
    const int l = lane & 15, hl = lane >> 4;
    const v2f* Bp = (const v2f*)Pfront;

    for (int tt = 0; tt < 5; ++tt) {
        int t = wv * 5 + tt;                       // 0..19
        const float* Bv; float* dst; int outd, n0;
        if (t < 16) { Bv = Bres;  dst = res_ws; outd = 256; n0 = t * 16; }
        else        { Bv = Binit; dst = a0_ws;  outd = 64;  n0 = (t - 16) * 16; }
        const int n = n0 + l;
        const int boff = t * (64 * 32) + lane;     // int, uniform base + lane

        v8f acc = {};
        for (int kk = 0; kk < 64; ++kk) {
            int k0 = kk * 4 + hl * 2;
            v2f a = *(const v2f*)&xs[l * 258 + k0];  // ds_load_b64
            v2f b = Bp[boff + kk * 32];              // global_load_b64 (saddr)
            acc = wmma_f32_4(a, b, acc);
        }
        float bias = Bv[n];
        for (int r = 0; r < 8; ++r) {
            int row = m0 + r + hl * 8;
            dst[row * outd + n] = leaky(acc[r] + bias);
        }
    }
}

// ---------------------------------------------------------------------------
// Kernel 2: LFA1. One wave per point i.
// ---------------------------------------------------------------------------
__global__ __launch_bounds__(256) void k_lfa1(
    const float* __restrict__ geom, const int* __restrict__ idx,
    const float* __restrict__ Pg1, const float* __restrict__ Bg,
    const float* __restrict__ a0_ws, float* __restrict__ a1_ws, int N)
{
    const int lane = threadIdx.x & 31, wv = threadIdx.x >> 5;
    const int i = __builtin_amdgcn_readfirstlane(blockIdx.x * 8 + wv);
    if (i >= N) return;                       // wave-uniform; EXEC stays full
    const int l = lane & 15, hl = lane >> 4;
    const float inv = 1.0f / 16.0f;

    // A fragment: rows = 16 neighbors, K = 4 geometry dims.
    v2f a = *(const v2f*)(geom + i * 64 + l * 4 + hl * 2);
    const v2f* Bp = (const v2f*)Pg1;

    for (int t = 0; t < 4; ++t) {
        int n = t * 16 + l;
        v2f b = Bp[t * 32 + lane];
        v8f c = {};
        c = wmma_f32_4(a, b, c);
        float bias = Bg[n];
        float s = 0.f;
        for (int r = 0; r < 8; ++r) s += leaky(c[r] + bias);
        s += __shfl_xor(s, 16, 32);           // combine row halves (M 0-7 / 8-15)
        if (hl == 0) a1_ws[i * 128 + n] = s * inv;
    }

    // Gather mean over neighbors (a0 is 12.8MB -> L2 hits). idx reads are
    // wave-uniform -> SMEM; row loads are uniform-base + lane offset.
    float s0 = 0.f, s1 = 0.f;
    const int* ip = idx + i * 16;
    for (int k = 0; k < 16; ++k) {
        int j = ip[k];
        int base = j * 64;
        s0 += a0_ws[base + lane];
        s1 += a0_ws[base + 32 + lane];
    }
    a1_ws[i * 128 + 64 + lane] = s0 * inv;
    a1_ws[i * 128 + 96 + lane] = s1 * inv;
}

// ---------------------------------------------------------------------------
// Kernel 3: fused LFA2 + final GEMM + residual add.
// 16 waves = 16 points per workgroup. Phase 1: a2 tile rows in LDS.
// Phase 2: wave wv -> output column tile wv*16, A-fragments via ds_load_b64.
// ---------------------------------------------------------------------------
__global__ __launch_bounds__(512) void k_lfa2_final(
    const float* __restrict__ geom, const int* __restrict__ idx,
    const float* __restrict__ Pg2, const float* __restrict__ Bg2,
    const float* __restrict__ a1_ws,
    const float* __restrict__ Pfin, const float* __restrict__ Bfin,
    const float* __restrict__ res_ws, float* __restrict__ out, int N)
{
    __shared__ float a2s[16 * 258];
    const int lane = threadIdx.x & 31, wv = threadIdx.x >> 5;
    const int m0 = blockIdx.x * 16;
    const int i  = __builtin_amdgcn_readfirstlane(m0 + wv);
    const int l = lane & 15, hl = lane >> 4;
    const float inv = 1.0f / 16.0f;

    // ---- Phase 1: LFA2 row for point i -> LDS row wv ----
    {
        v2f a = *(const v2f*)(geom + i * 64 + l * 4 + hl * 2);
        const v2f* Bp = (const v2f*)Pg2;
        for (int t = 0; t < 8; ++t) {
            int n = t * 16 + l;
            v2f b = Bp[t * 32 + lane];
            v8f c = {};
            c = wmma_f32_4(a, b, c);
            float bias = Bg2[n];
            float s = 0.f;
            for (int r = 0; r < 8; ++r) s += leaky(c[r] + bias);
            s += __shfl_xor(s, 16, 32);
            if (hl == 0) a2s[wv * 258 + n] = s * inv;
        }
        float s0 = 0.f, s1 = 0.f, s2 = 0.f, s3 = 0.f;
        const int* ip = idx + i * 16;
        for (int k = 0; k < 16; ++k) {
            int j = ip[k];
            int base = j * 128;                      // 25.6MB -> L2 hits
            s0 += a1_ws[base + lane];
            s1 += a1_ws[base + 32 + lane];
            s2 += a1_ws[base + 64 + lane];
            s3 += a1_ws[base + 96 + lane];
        }
        a2s[wv * 258 + 128 + lane] = s0 * inv;
        a2s[wv * 258 + 160 + lane] = s1 * inv;
        a2s[wv * 258 + 192 + lane] = s2 * inv;
        a2s[wv * 258 + 224 + lane] = s3 * inv;
    }
    __syncthreads();

    // ---- Phase 2: final GEMM column tile, 64 WMMAs per wave ----
    {
        const int n = wv * 16 + l;
        const v2f* Bp = (const v2f*)Pfin;
        const int boff = wv * (64 * 32) + lane;
        v8f acc = {};
        for (int kk = 0; kk < 64; ++kk) {
            int k0 = kk * 4 + hl * 2;
            v2f a = *(const v2f*)&a2s[l * 258 + k0];  // ds_load_b64
            v2f b = Bp[boff + kk * 32];               // global_load_b64 (saddr)
            acc = wmma_f32_4(a, b, acc);
        }
        float bias = Bfin[n];
        for (int r = 0; r < 8; ++r) {
            int row = m0 + r + hl * 8;
            float v = leaky(acc[r] + bias);
            out[row * 256 + n] = v + res_ws[row * 256 + n];
        }
    }
}

// ---------------------------------------------------------------------------
extern "C" void kernel_launch(void* const* d_in, const int* in_sizes, int n_in,
                              void* d_out, int out_size, void* d_ws, size_t ws_size,
                              hipStream_t stream) {
    const float* X     = (const float*)d_in[0];   // [N,256]
    const float* geom  = (const float*)d_in[1];   // [N,16,4]
    const int*   idx   = (const int*)  d_in[2];   // [N,16]
    const float* Wres  = (const float*)d_in[3];   // [256,256]
    const float* Bres  = (const float*)d_in[4];   // [256]
    const float* Winit = (const float*)d_in[5];   // [256,64]
    const float* Binit = (const float*)d_in[6];   // [64]
    const float* Wg1   = (const float*)d_in[7];   // [4,64]
    const float* Bg1   = (const float*)d_in[8];   // [64]
    const float* Wg2   = (const float*)d_in[9];   // [4,128]
    const float* Bg2   = (const float*)d_in[10];  // [128]
    const float* Wfin  = (const float*)d_in[11];  // [256,256]
    const float* Bfin  = (const float*)d_in[12];  // [256]

    const int N = in_sizes[0] / 256;              // 50000 (divisible by 16)

    float* ws      = (float*)d_ws;
    float* res_ws  = ws;                          // N*256 floats
    float* a0_ws   = res_ws + (size_t)N * 256;    // N*64  floats
    float* a1_ws   = a0_ws  + (size_t)N * 64;     // N*128 floats
    float* Pfront  = a1_ws  + (size_t)N * 128;    // NFRONT*2 floats
    float* Pfin    = Pfront + (size_t)NFRONT * 2; // NFIN*2 floats
    float* Pg1     = Pfin   + (size_t)NFIN * 2;   // NG1*2 floats
    float* Pg2     = Pg1    + (size_t)NG1 * 2;    // NG2*2 floats

    const int mtiles = N / 16;                    // 3125
    const int npack  = NFRONT + NFIN + NG1 + NG2;

    k_pack<<<(npack + 255) / 256, 256, 0, stream>>>(Wres, Winit, Wg1, Wg2, Wfin,
                                                    Pfront, Pg1, Pg2, Pfin);
    k_front<<<mtiles, 128, 0, stream>>>(X, Pfront, Bres, Binit,
                                        res_ws, a0_ws, N);
    k_lfa1<<<(N + 7) / 8, 256, 0, stream>>>(geom, idx, Pg1, Bg1,
                                            a0_ws, a1_ws, N);
    k_lfa2_final<<<mtiles, 512, 0, stream>>>(geom, idx, Pg2, Bg2, a1_ws,
                                             Pfin, Bfin, res_ws,
                                             (float*)d_out, N);
}